// MMCL_52029233824081
// MI455X (gfx1250) — compile-verified
//
#include <hip/hip_runtime.h>
#include <cstdint>

// Problem constants (match reference)
#define M_ROWS   4096
#define N_COLS   16384
#define K_HARD   163          // int(0.01 * (16384-1))
#define DELTA_W  5.0f
#define BLOCK    256
#define NWAVES   (BLOCK / 32)
#define CHUNK    2048         // floats per TDM chunk (8 KB)
#define NCHUNK   (N_COLS / CHUNK)

typedef unsigned int uint32;
typedef __attribute__((ext_vector_type(4))) unsigned int v4u;
typedef __attribute__((ext_vector_type(8))) int          v8i;
typedef __attribute__((ext_vector_type(4))) int          v4i;

// Order-preserving float <-> uint mapping (larger float => larger uint)
__device__ __forceinline__ uint32 f2ord(float x) {
  uint32 s = __float_as_uint(x);
  return (s & 0x80000000u) ? ~s : (s | 0x80000000u);
}
__device__ __forceinline__ float ord2f(uint32 u) {
  uint32 s = (u & 0x80000000u) ? (u ^ 0x80000000u) : ~u;
  return __uint_as_float(s);
}
__device__ __forceinline__ float softplusf(float x) {
  // numerically stable: max(x,0) + log1p(exp(-|x|))
  return fmaxf(x, 0.0f) + log1pf(expf(-fabsf(x)));
}

// Issue one TDM DMA: CHUNK contiguous f32 from global -> LDS.
// D# layout per CDNA5 ISA ch.8 (groups 0/1; groups 2/3 zero => <=2D tensor).
// 6-arg builtin form: (g0, g1, g2, g3, extra_g, cpol).
__device__ __forceinline__ void tdm_load_chunk(const float* gsrc, uint32 lds_off) {
  unsigned long long ga = (unsigned long long)(uintptr_t)gsrc;
  v4u g0;
  g0[0] = 1u;                                            // count=1 (valid), no gather
  g0[1] = lds_off;                                       // lds_addr (bytes)
  g0[2] = (uint32)(ga & 0xFFFFFFFFull);                  // global_addr[31:0]
  g0[3] = (uint32)((ga >> 32) & 0x1FFFFFFull)            // global_addr[56:32]
        | (2u << 30);                                    // type=2 ("image")
  v8i g1;
  g1[0] = (2 << 16);                                     // workgroup_mask=0, data_size=2 (4B)
  g1[1] = (int)((CHUNK & 0xFFFF) << 16);                 // tensor_dim0[15:0]  @ bits 63:48
  g1[2] = (int)(CHUNK >> 16);                            // tensor_dim0[31:16] @ bits 79:64
  g1[3] = (int)((CHUNK & 0xFFFF) << 16);                 // tile_dim0 @ bits 127:112
  g1[4] = 0;                                             // tile_dim1/2 = 0 (unused)
  g1[5] = (int)CHUNK;                                    // tensor_dim0_stride lo32
  g1[6] = 0;
  g1[7] = 0;
  v4i gz  = {0, 0, 0, 0};
  v8i gz8 = {0, 0, 0, 0, 0, 0, 0, 0};
  __builtin_amdgcn_tensor_load_to_lds(g0, g1, gz, gz, gz8, 0 /*cpol*/);
}

__global__ __launch_bounds__(BLOCK) void topk_bce_rows(
    const float* __restrict__ inputs,
    const int*   __restrict__ targets,
    float*       __restrict__ rowout)
{
  __shared__ alignas(16) float buf[2][CHUNK];  // 16 KB TDM double buffer
  __shared__ uint32 histw[NWAVES][256];        // per-wave replicated pass-0 hist (8 KB)
  __shared__ uint32 hist[256];                 // combined / refinement hist (1 KB)
  __shared__ float  red[BLOCK];
  __shared__ uint32 sh_prefix, sh_k;
  __shared__ float  sh_pos;

  const int    row  = blockIdx.x;
  const int    tid  = threadIdx.x;
  const int    wid  = tid >> 5;
  const float* rowp = inputs + (size_t)row * N_COLS;
  const int    tcol = targets[row];

  // Wave-uniform predicate -> scalar branch (TDM ignores EXEC, must truly skip)
  const bool wave0 = (__builtin_amdgcn_readfirstlane((int)(threadIdx.x >> 5)) == 0);

  for (int b = tid; b < NWAVES * 256; b += BLOCK) (&histw[0][0])[b] = 0u;
  if (tid == 0) sh_pos = 0.0f;
  __syncthreads();

  // ---------------- Pass 0: TDM-streamed top-8-bit histogram ----------------
  // Per-wave replicated bins: Gaussian data concentrates ~45% of mass in one
  // exponent bucket; replication x8 cuts the same-address LDS RMW chain 8x.
  if (wave0) {
    tdm_load_chunk(rowp, (uint32)(size_t)(void*)&buf[0][0]);
  }
  for (int c = 0; c < NCHUNK; ++c) {
    if (wave0) {
      if (c + 1 < NCHUNK) {
        tdm_load_chunk(rowp + (size_t)(c + 1) * CHUNK,
                       (uint32)(size_t)(void*)&buf[(c + 1) & 1][0]);
        __builtin_amdgcn_s_wait_tensorcnt(1);  // oldest (chunk c) complete
      } else {
        __builtin_amdgcn_s_wait_tensorcnt(0);
      }
    }
    __syncthreads();                           // publish LDS tile to all waves
    const float4* cb4  = reinterpret_cast<const float4*>(&buf[c & 1][0]);
    const int     gbase = c * CHUNK;
    for (int j = tid; j < CHUNK / 4; j += BLOCK) {
      float4 v = cb4[j];
      float xs[4] = {v.x, v.y, v.z, v.w};
      int gidx = gbase + j * 4;
#pragma unroll
      for (int e = 0; e < 4; ++e) {
        if (gidx + e == tcol) { sh_pos = xs[e]; continue; }  // exclude positive
        atomicAdd(&histw[wid][f2ord(xs[e]) >> 24], 1u);
      }
    }
    __syncthreads();                           // reads done before buffer refill
  }

  // Combine wave copies, then serial 256-bin suffix scan by thread 0
  {
    uint32 s = 0u;
#pragma unroll
    for (int w = 0; w < NWAVES; ++w) s += histw[w][tid];
    hist[tid] = s;
  }
  __syncthreads();
  if (tid == 0) {
    uint32 need = K_HARD, acc = 0u;
    int b;
    for (b = 255; b > 0; --b) {
      uint32 h = hist[b];
      if (acc + h >= need) break;
      acc += h;
    }
    sh_prefix = (uint32)b;
    sh_k      = need - acc;                    // rank within chosen bucket
  }
  __syncthreads();

  // ------------- Passes 1..3: radix refinement (row is L2-hot) -------------
  // Only ~64-200 elements match the prefix per pass => negligible contention.
  const float4* rowp4 = reinterpret_cast<const float4*>(rowp);
  for (int p = 1; p <= 3; ++p) {
    hist[tid] = 0u;
    __syncthreads();
    const uint32 prefix = sh_prefix;
    const int shP = 32 - 8 * p;                // prefix-compare shift (24/16/8)
    const int shB = 24 - 8 * p;                // bucket shift (16/8/0)
    for (int j = tid; j < N_COLS / 4; j += BLOCK) {
      float4 v = rowp4[j];
      float xs[4] = {v.x, v.y, v.z, v.w};
      int gidx = j * 4;
#pragma unroll
      for (int e = 0; e < 4; ++e) {
        if (gidx + e == tcol) continue;
        uint32 u = f2ord(xs[e]);
        if ((u >> shP) == prefix)
          atomicAdd(&hist[(u >> shB) & 255u], 1u);
      }
    }
    __syncthreads();
    if (tid == 0) {
      uint32 need = sh_k, acc = 0u;
      int b;
      for (b = 255; b > 0; --b) {
        uint32 h = hist[b];
        if (acc + h >= need) break;
        acc += h;
      }
      sh_prefix = (sh_prefix << 8) | (uint32)b;
      sh_k      = need - acc;
    }
    __syncthreads();
  }

  // -------- Final pass: sum softplus over strict-greater + tie credit -------
  const uint32 thr  = sh_prefix;               // exact 163rd-largest (ordered bits)
  const uint32 krem = sh_k;                    // how many threshold-valued to count
  float lsum = 0.0f;
  for (int j = tid; j < N_COLS / 4; j += BLOCK) {
    float4 v = rowp4[j];
    float xs[4] = {v.x, v.y, v.z, v.w};
    int gidx = j * 4;
#pragma unroll
    for (int e = 0; e < 4; ++e) {
      if (gidx + e == tcol) continue;
      if (f2ord(xs[e]) > thr) lsum += softplusf(xs[e]);
    }
  }
  red[tid] = lsum;
  __syncthreads();
  for (int s = BLOCK / 2; s > 0; s >>= 1) {    // deterministic tree reduction
    if (tid < s) red[tid] += red[tid + s];
    __syncthreads();
  }
  if (tid == 0) {
    float sneg = red[0] + (float)krem * softplusf(ord2f(thr));
    float lpos = softplusf(-sh_pos);
    rowout[row] = DELTA_W * lpos + sneg * (1.0f / (float)K_HARD);
  }
}

__global__ __launch_bounds__(BLOCK) void reduce_rows(
    const float* __restrict__ rowout, float* __restrict__ out)
{
  __shared__ float red[BLOCK];
  float s = 0.0f;
  for (int i = threadIdx.x; i < M_ROWS; i += BLOCK) s += rowout[i];
  red[threadIdx.x] = s;
  __syncthreads();
  for (int st = BLOCK / 2; st > 0; st >>= 1) {
    if (threadIdx.x < st) red[threadIdx.x] += red[threadIdx.x + st];
    __syncthreads();
  }
  if (threadIdx.x == 0) out[0] = red[0] * (1.0f / (float)M_ROWS);
}

extern "C" void kernel_launch(void* const* d_in, const int* in_sizes, int n_in,
                              void* d_out, int out_size, void* d_ws, size_t ws_size,
                              hipStream_t stream)
{
  (void)in_sizes; (void)n_in; (void)out_size; (void)ws_size;
  const float* inputs  = (const float*)d_in[0];
  const int*   targets = (const int*)d_in[1];
  float*       rowbuf  = (float*)d_ws;            // M_ROWS floats of scratch

  topk_bce_rows<<<M_ROWS, BLOCK, 0, stream>>>(inputs, targets, rowbuf);
  reduce_rows<<<1, BLOCK, 0, stream>>>(rowbuf, (float*)d_out);
}